// NormalizingFlow_51384988729483
// MI455X (gfx1250) — compile-verified
//
#include <hip/hip_runtime.h>
#include <cmath>

typedef float v2f __attribute__((ext_vector_type(2)));
typedef float v8f __attribute__((ext_vector_type(8)));

#define N_SAMPLES 65536
#define DIM 512
#define NFLOW 32

// ---------------- prep kernel: u_hat, G = Uhat @ W^T, wuh = diag(G) ----------------
__global__ void planar_prep(const float* __restrict__ u,
                            const float* __restrict__ w,
                            float* __restrict__ uhat,
                            float* __restrict__ G,
                            float* __restrict__ wuh) {
  __shared__ float s_corr[NFLOW];
  __shared__ float s_uhat[NFLOW * DIM];
  int tid = threadIdx.x;
  if (tid < NFLOW) {
    float wtu = 0.f, wn2 = 0.f;
    const float* wr = w + tid * DIM;
    const float* ur = u + tid * DIM;
    for (int d = 0; d < DIM; ++d) { wtu = fmaf(wr[d], ur[d], wtu); wn2 = fmaf(wr[d], wr[d], wn2); }
    float m = -1.0f + log1pf(expf(wtu));
    // u_hat = u + (m - wtu) * w / ||w||^2  when wtu < 1, else u
    s_corr[tid] = (wtu < 1.0f) ? ((m - wtu) / wn2) : 0.0f;
  }
  __syncthreads();
  for (int idx = tid; idx < NFLOW * DIM; idx += blockDim.x) {
    int i = idx >> 9;  // / DIM
    float uh = u[idx] + s_corr[i] * w[idx];
    s_uhat[idx] = uh;
    uhat[idx] = uh;
  }
  __syncthreads();
  for (int e = tid; e < NFLOW * NFLOW; e += blockDim.x) {
    int j = e >> 5, k = e & 31;
    const float* uj = s_uhat + j * DIM;
    const float* wk = w + k * DIM;
    float acc = 0.f;
    for (int d = 0; d < DIM; ++d) acc = fmaf(uj[d], wk[d], acc);
    G[e] = acc;
    if (j == k) wuh[j] = acc;  // w_k . u_hat_k
  }
}

// ---------------- fused main kernel ----------------
// Phase 1: L0 = X @ W^T via v_wmma_f32_16x16x4_f32  (per wave: 32 rows, 2x2 C tiles)
// Phase 2: per-lane triangular recurrence: t_k = tanh(L0_k + b_k + sum_{j<k} t_j G[j][k])
// Phase 3: Out = X + T @ Uhat via WMMA
__global__ __launch_bounds__(256)
void planar_main(const float* __restrict__ x,
                 const float* __restrict__ wmat,
                 const float* __restrict__ bvec,
                 const float* __restrict__ uhat,
                 const float* __restrict__ G,
                 const float* __restrict__ wuh,
                 float* __restrict__ out,
                 float* __restrict__ logdet) {
  __shared__ float Ts[256 * 34];     // padded rows: 34 (17 odd mod 32 -> conflict-free)
  __shared__ float Gs[NFLOW * NFLOW];
  __shared__ float bs[NFLOW];
  __shared__ float wuhs[NFLOW];

  const int tid  = threadIdx.x;
  const int lane = tid & 31;
  const int wv   = tid >> 5;                 // wave id, rows [wv*32, wv*32+32)
  const int mrow = lane & 15;                // M (or N) index within fragment
  const int kadd = (lane >> 4) << 1;         // 0 or 2 : K sub-offset per ISA A/B layout
  const int crow = (lane >> 4) << 3;         // 0 or 8 : C row offset
  const int ccol = lane & 15;                // C col
  const int blockRow = blockIdx.x * 256;
  const int r0w = blockRow + wv * 32;

  for (int e = tid; e < NFLOW * NFLOW; e += 256) Gs[e] = G[e];
  if (tid < NFLOW) { bs[tid] = bvec[tid]; wuhs[tid] = wuh[tid]; }
  __syncthreads();

  // ---- Phase 1: lin = X @ W^T ----
  v8f c00 = {}, c01 = {}, c10 = {}, c11 = {};
  const float* xA0 = x + (size_t)(r0w + mrow) * DIM + kadd;
  const float* xA1 = xA0 + 16 * DIM;
  const float* wB0 = wmat + (size_t)mrow * DIM + kadd;        // flows 0..15
  const float* wB1 = wB0 + 16 * DIM;                          // flows 16..31
  for (int k0 = 0; k0 < DIM; k0 += 4) {
    v2f a0 = *(const v2f*)(xA0 + k0);
    v2f a1 = *(const v2f*)(xA1 + k0);
    v2f b0 = *(const v2f*)(wB0 + k0);
    v2f b1 = *(const v2f*)(wB1 + k0);
    c00 = __builtin_amdgcn_wmma_f32_16x16x4_f32(false, a0, false, b0, (short)0, c00, false, false);
    c01 = __builtin_amdgcn_wmma_f32_16x16x4_f32(false, a0, false, b1, (short)0, c01, false, false);
    c10 = __builtin_amdgcn_wmma_f32_16x16x4_f32(false, a1, false, b0, (short)0, c10, false, false);
    c11 = __builtin_amdgcn_wmma_f32_16x16x4_f32(false, a1, false, b1, (short)0, c11, false, false);
  }
  #pragma unroll
  for (int v = 0; v < 8; ++v) {
    int rbase = wv * 32 + v + crow;
    Ts[(rbase     ) * 34 + ccol     ] = c00[v];
    Ts[(rbase     ) * 34 + 16 + ccol] = c01[v];
    Ts[(rbase + 16) * 34 + ccol     ] = c10[v];
    Ts[(rbase + 16) * 34 + 16 + ccol] = c11[v];
  }
  __syncthreads();

  // ---- Phase 2: per-lane recurrence over the 32 flows (lane owns row == tid) ----
  {
    float lin[NFLOW];
    #pragma unroll
    for (int k = 0; k < NFLOW; ++k) lin[k] = Ts[tid * 34 + k] + bs[k];
    float ld = 0.f;
    #pragma unroll
    for (int k = 0; k < NFLOW; ++k) {
      float t = tanhf(lin[k]);
      ld += logf(fabsf(fmaf(1.0f - t * t, wuhs[k], 1.0f)));
      Ts[tid * 34 + k] = t;  // overwrite lin with t in-place
      #pragma unroll
      for (int j = k + 1; j < NFLOW; ++j) lin[j] = fmaf(t, Gs[k * NFLOW + j], lin[j]);
    }
    logdet[blockRow + tid] = ld;
  }
  __syncthreads();

  // ---- Phase 3: Out = X + T @ Uhat ----
  const int arow0 = (wv * 32 + mrow) * 34 + kadd;
  #pragma unroll 1
  for (int nt = 0; nt < DIM / 16; ++nt) {
    const int ncol = nt * 16 + mrow;
    #pragma unroll
    for (int mt = 0; mt < 2; ++mt) {
      v8f c = {};
      #pragma unroll
      for (int k0 = 0; k0 < NFLOW; k0 += 4) {
        v2f a;
        a.x = Ts[arow0 + mt * 16 * 34 + k0];
        a.y = Ts[arow0 + mt * 16 * 34 + k0 + 1];
        v2f bb;
        bb.x = uhat[(k0 + kadd)     * DIM + ncol];
        bb.y = uhat[(k0 + kadd + 1) * DIM + ncol];
        c = __builtin_amdgcn_wmma_f32_16x16x4_f32(false, a, false, bb, (short)0, c, false, false);
      }
      #pragma unroll
      for (int v = 0; v < 8; ++v) {
        size_t row = (size_t)(r0w + mt * 16 + v + crow);
        size_t idx = row * DIM + nt * 16 + ccol;
        out[idx] = x[idx] + c[v];
      }
    }
  }
}

extern "C" void kernel_launch(void* const* d_in, const int* in_sizes, int n_in,
                              void* d_out, int out_size, void* d_ws, size_t ws_size,
                              hipStream_t stream) {
  const float* x = (const float*)d_in[0];
  const float* u = (const float*)d_in[1];
  const float* w = (const float*)d_in[2];
  const float* b = (const float*)d_in[3];

  float* outp    = (float*)d_out;                       // [N, D]
  float* logdetp = outp + (size_t)N_SAMPLES * DIM;      // [N]

  float* uhat = (float*)d_ws;                           // NFLOW*DIM floats
  float* G    = uhat + NFLOW * DIM;                     // NFLOW*NFLOW
  float* wuh  = G + NFLOW * NFLOW;                      // NFLOW

  planar_prep<<<1, 256, 0, stream>>>(u, w, uhat, G, wuh);
  planar_main<<<N_SAMPLES / 256, 256, 0, stream>>>(x, w, b, uhat, G, wuh, outp, logdetp);
}